// GraUNet_43026982371447
// MI455X (gfx1250) — compile-verified
//
#include <hip/hip_runtime.h>
#include <hip/hip_bf16.h>

// ---------------------------------------------------------------------------
// Graph U-Net (GraUNet) for MI455X / gfx1250.
//  * Dense GEMMs: V_WMMA_F32_16X16X4_F32, 16x64 strip per wave (4 acc tiles
//    share one A fragment).  W panel staged in LDS in k-pair-interleaved
//    layout so every B fragment is one aligned ds_load_b64 (no repack moves).
//  * GCN diffusion: deterministic CSR gather (per-level CSR with edge-id
//    sorted buckets), fused hop kernel, zero fp atomics in the hot path.
//  * TopK pooling fully on device (single-block bisection threshold).
// ---------------------------------------------------------------------------

typedef __attribute__((ext_vector_type(2))) float v2f;
typedef __attribute__((ext_vector_type(8))) float v8f;

#define N0 50000
#define N1 25000
#define N2 12500
#define NE 800000
#define CH 128
#define CCAT 256
#define COUT 64

static inline int divup(int a, int b) { return (a + b - 1) / b; }

// ------------------------------ elementwise --------------------------------

__global__ void fill_f32(float* p, float v, int n) {
    int i = blockIdx.x * blockDim.x + threadIdx.x;
    if (i < n) p[i] = v;
}
__global__ void fill_i32(int* p, int v, int n) {
    int i = blockIdx.x * blockDim.x + threadIdx.x;
    if (i < n) p[i] = v;
}
__global__ void copy_f32(float* dst, const float* __restrict__ src, int n) {
    int i = blockIdx.x * blockDim.x + threadIdx.x;
    if (i < n) dst[i] = src[i];
}
__global__ void relu_f32(float* p, int n) {
    int i = blockIdx.x * blockDim.x + threadIdx.x;
    if (i < n) p[i] = fmaxf(p[i], 0.0f);
}

// ------------------------------ WMMA GEMM ----------------------------------
// Out[M x N] = A[M x K] * W[K x N].  Block = 256 threads (8 waves).
// Block tile = 128 rows x 64 cols; each wave owns a 16x64 strip (4 WMMA
// accumulators reusing one A fragment per k-step).
// LDS layout: sW[(k>>1)*128 + c*2 + (k&1)] -> (W[k][c], W[k+1][c]) adjacent,
// so a B fragment is a single aligned v2f load.

template <int KMAX>
__global__ void gemm_wmma_f32(const float* __restrict__ A,
                              const float* __restrict__ W,
                              float* __restrict__ Out,
                              int M, int K, int N) {
    __shared__ float sW[KMAX * 64];
    const int tid  = threadIdx.x;
    const int col0 = blockIdx.y * 64;
    const int row0 = blockIdx.x * 128;

    const int kn = K * 64;
    for (int idx = tid; idx < kn; idx += 256) {
        int k = idx >> 6, c = idx & 63;
        sW[(k >> 1) * 128 + (c << 1) + (k & 1)] = W[k * N + col0 + c];
    }
    __syncthreads();

    const int wave  = tid >> 5;
    const int lane  = tid & 31;
    const int nloc  = lane & 15;
    const int khalf = lane >> 4;            // 0 or 1
    const int m0    = row0 + wave * 16;
    int mrow = m0 + nloc;                   // A row this lane loads
    if (mrow >= M) mrow = M - 1;            // clamp (stores are guarded)

    v8f acc0 = {}, acc1 = {}, acc2 = {}, acc3 = {};
    for (int k = 0; k < K; k += 4) {
        v2f a = *(const v2f*)(A + (size_t)mrow * K + k + khalf * 2);
        // pair-row index for this lane-half: rows (k + khalf*2, k + khalf*2 + 1)
        const float* bp = &sW[((k >> 1) + khalf) * 128 + (nloc << 1)];
        v2f b;
        b = *(const v2f*)(bp + 0);
        acc0 = __builtin_amdgcn_wmma_f32_16x16x4_f32(false, a, false, b, (short)0, acc0, false, false);
        b = *(const v2f*)(bp + 32);
        acc1 = __builtin_amdgcn_wmma_f32_16x16x4_f32(false, a, false, b, (short)0, acc1, false, false);
        b = *(const v2f*)(bp + 64);
        acc2 = __builtin_amdgcn_wmma_f32_16x16x4_f32(false, a, false, b, (short)0, acc2, false, false);
        b = *(const v2f*)(bp + 96);
        acc3 = __builtin_amdgcn_wmma_f32_16x16x4_f32(false, a, false, b, (short)0, acc3, false, false);
    }

    v8f* accs[4] = {&acc0, &acc1, &acc2, &acc3};
    #pragma unroll
    for (int t = 0; t < 4; ++t) {
        int ncol = col0 + t * 16 + nloc;
        if (ncol < N) {
            #pragma unroll
            for (int i = 0; i < 8; ++i) {
                int m = m0 + i + 8 * khalf;
                if (m < M) Out[(size_t)m * N + ncol] = (*accs[t])[i];
            }
        }
    }
}

// --------------------- per-level graph preprocessing -----------------------

__global__ void deg_kernel(const int* __restrict__ er, const float* __restrict__ ew,
                           float* deg, int E) {
    int e = blockIdx.x * blockDim.x + threadIdx.x;
    if (e < E) atomicAdd(&deg[er[e]], ew[e]);   // ew in {0,1}: exact, order-free
}
__global__ void dinv_kernel(float* deg, int n) {  // deg -> rsqrt(deg + 1)
    int i = blockIdx.x * blockDim.x + threadIdx.x;
    if (i < n) deg[i] = rsqrtf(deg[i] + 1.0f);
}
__global__ void norm_kernel(const int* __restrict__ er, const int* __restrict__ ec,
                            const float* __restrict__ ew, const float* __restrict__ dinv,
                            float* norm, int E) {
    int e = blockIdx.x * blockDim.x + threadIdx.x;
    if (e < E) norm[e] = dinv[er[e]] * ew[e] * dinv[ec[e]];
}
__global__ void rowcnt_kernel(const int* __restrict__ er, const float* __restrict__ ew,
                              int* cnt, int E) {
    int e = blockIdx.x * blockDim.x + threadIdx.x;
    if (e < E && ew[e] != 0.0f) atomicAdd(&cnt[er[e]], 1);
}
// single-block exclusive scan: rp[0]=0, rp[i+1]=sum cnt[0..i]
__global__ void scan_kernel(const int* __restrict__ cnt, int* rp, int n) {
    __shared__ int sdata[1024];
    __shared__ int carry;
    if (threadIdx.x == 0) carry = 0;
    __syncthreads();
    for (int base = 0; base < n; base += 1024) {
        int i = base + threadIdx.x;
        int v = (i < n) ? cnt[i] : 0;
        sdata[threadIdx.x] = v;
        __syncthreads();
        for (int off = 1; off < 1024; off <<= 1) {
            int t = (threadIdx.x >= off) ? sdata[threadIdx.x - off] : 0;
            __syncthreads();
            sdata[threadIdx.x] += t;
            __syncthreads();
        }
        int incl = sdata[threadIdx.x] + carry;
        if (i < n) rp[i + 1] = incl;
        __syncthreads();
        if (threadIdx.x == 1023) carry = incl;
        __syncthreads();
    }
    if (threadIdx.x == 0) rp[0] = 0;
}
__global__ void csr_fill_kernel(const int* __restrict__ er, const float* __restrict__ ew,
                                const int* __restrict__ rp, int* pos, int* eid, int E) {
    int e = blockIdx.x * blockDim.x + threadIdx.x;
    if (e >= E || ew[e] == 0.0f) return;
    int r = er[e];
    int p = atomicAdd(&pos[r], 1);
    eid[rp[r] + p] = e;
}
// sort each row bucket ascending by edge id -> fixed summation order
__global__ void sort_rows_kernel(const int* __restrict__ rp, int* eid, int n) {
    int r = blockIdx.x * blockDim.x + threadIdx.x;
    if (r >= n) return;
    int b = rp[r], e = rp[r + 1];
    for (int i = b + 1; i < e; ++i) {
        int v = eid[i];
        int j = i - 1;
        while (j >= b && eid[j] > v) { eid[j + 1] = eid[j]; --j; }
        eid[j + 1] = v;
    }
}
__global__ void csr_mat_kernel(const int* __restrict__ eid, const int* __restrict__ ec,
                               const float* __restrict__ nrm,
                               int* ccol, float* cw, const int* __restrict__ rp,
                               int n, int E) {
    int j = blockIdx.x * blockDim.x + threadIdx.x;
    if (j >= E || j >= rp[n]) return;
    int e = eid[j];
    ccol[j] = ec[e];
    cw[j]   = nrm[e];
}

// -------------------------- fused diffusion hop ----------------------------
// hnew[r,:] = sum_j cw[j] * h[ccol[j],:] + dinv[r]^2 * h[r,:];  out += hnew
// C=128: one wave per row (32 lanes x float4), uniform loop per wave.
__global__ void hop_kernel(const int* __restrict__ rp, const int* __restrict__ ccol,
                           const float* __restrict__ cw, const float* __restrict__ dinv,
                           const float* __restrict__ h, float* __restrict__ hnew,
                           float* __restrict__ out, int n, int cshift) {
    int gid = blockIdx.x * blockDim.x + threadIdx.x;
    int gshift = cshift - 2;
    int r = gid >> gshift;
    if (r >= n) return;
    int f4 = (gid & ((1 << gshift) - 1)) << 2;
    int jb = rp[r], je = rp[r + 1];
    float4 acc = make_float4(0.f, 0.f, 0.f, 0.f);
    for (int j = jb; j < je; ++j) {
        int c = ccol[j];
        float w = cw[j];
        const float4 v = *(const float4*)(h + ((size_t)c << cshift) + f4);
        acc.x += w * v.x; acc.y += w * v.y; acc.z += w * v.z; acc.w += w * v.w;
    }
    float d = dinv[r];
    float s = d * d;
    size_t off = ((size_t)r << cshift) + f4;
    const float4 hp = *(const float4*)(h + off);
    float4 v;
    v.x = acc.x + s * hp.x; v.y = acc.y + s * hp.y;
    v.z = acc.z + s * hp.z; v.w = acc.w + s * hp.w;
    *(float4*)(hnew + off) = v;
    float4 o = *(float4*)(out + off);
    o.x += v.x; o.y += v.y; o.z += v.z; o.w += v.w;
    *(float4*)(out + off) = o;
}

// ------------------------------ TopK pooling -------------------------------

__global__ void pwnorm_kernel(const float* __restrict__ pw, float* invn, int C) {
    __shared__ float red[128];
    float v = (threadIdx.x < C) ? pw[threadIdx.x] : 0.0f;
    red[threadIdx.x] = v * v;
    __syncthreads();
    for (int s = 64; s > 0; s >>= 1) {
        if (threadIdx.x < s) red[threadIdx.x] += red[threadIdx.x + s];
        __syncthreads();
    }
    if (threadIdx.x == 0) *invn = rsqrtf(red[0]);
}

__global__ void score_kernel(const float* __restrict__ x, const float* __restrict__ pw,
                             const float* __restrict__ invn,
                             float* score, unsigned* key, int n, int C) {
    int gid  = blockIdx.x * blockDim.x + threadIdx.x;
    int row  = gid >> 5;
    int lane = gid & 31;
    if (row >= n) return;
    float s = 0.0f;
    for (int f = lane; f < C; f += 32) s += x[(size_t)row * C + f] * pw[f];
    for (int off = 16; off > 0; off >>= 1) s += __shfl_down(s, off, 32);
    if (lane == 0) {
        float sc = tanhf(s * (*invn));
        score[row] = sc;
        unsigned u = __float_as_uint(sc);
        key[row] = (u & 0x80000000u) ? ~u : (u | 0x80000000u);
    }
}

// single-block bisection: largest t with count(key >= t) >= k
__global__ void topk_thresh_kernel(const unsigned* __restrict__ key,
                                   int n, int k, unsigned* thresh) {
    __shared__ int red[1024];
    unsigned long long lo = 0ull, hi = 0xFFFFFFFFull;
    for (int it = 0; it < 33 && lo < hi; ++it) {
        unsigned long long mid = (lo + hi + 1ull) >> 1;
        unsigned m32 = (unsigned)mid;
        int c = 0;
        for (int i = threadIdx.x; i < n; i += blockDim.x) c += (key[i] >= m32) ? 1 : 0;
        red[threadIdx.x] = c;
        __syncthreads();
        for (int s = 512; s > 0; s >>= 1) {
            if (threadIdx.x < s) red[threadIdx.x] += red[threadIdx.x + s];
            __syncthreads();
        }
        int total = red[0];
        __syncthreads();
        if (total >= k) lo = mid; else hi = mid - 1ull;
    }
    if (threadIdx.x == 0) *thresh = (unsigned)lo;
}

__global__ void select_topk_kernel(const unsigned* __restrict__ key,
                                   const unsigned* __restrict__ thresh,
                                   int* counter, int* perm, int n, int k) {
    int i = blockIdx.x * blockDim.x + threadIdx.x;
    if (i >= n) return;
    if (key[i] >= *thresh) {
        int p = atomicAdd(counter, 1);
        if (p < k) perm[p] = i;
    }
}
__global__ void build_nm_kernel(const int* __restrict__ perm, int* nm, int k) {
    int j = blockIdx.x * blockDim.x + threadIdx.x;
    if (j < k) nm[perm[j]] = j;
}
__global__ void gather_gate_kernel(const float* __restrict__ x, const int* __restrict__ perm,
                                   const float* __restrict__ score, float* x2,
                                   int k, int cshift) {
    int gid = blockIdx.x * blockDim.x + threadIdx.x;
    int j = gid >> cshift;
    if (j >= k) return;
    int f = gid & ((1 << cshift) - 1);
    int src = perm[j];
    x2[gid] = x[((size_t)src << cshift) + f] * score[src];
}
__global__ void relabel_kernel(const int* __restrict__ er, const int* __restrict__ ec,
                               const float* __restrict__ ew, const int* __restrict__ nm,
                               int* er2, int* ec2, float* ew2, int E) {
    int e = blockIdx.x * blockDim.x + threadIdx.x;
    if (e >= E) return;
    int r = nm[er[e]], c = nm[ec[e]];
    bool kept = (r >= 0) && (c >= 0);
    er2[e] = kept ? r : 0;
    ec2[e] = kept ? c : 0;
    ew2[e] = kept ? ew[e] : 0.0f;
}

// up-path: out[perm[j], :] += xsmall[j, :]   (perm entries unique -> plain add)
__global__ void scatter_add_kernel(float* out, const float* __restrict__ xsmall,
                                   const int* __restrict__ perm, int k, int cshift) {
    int gid = blockIdx.x * blockDim.x + threadIdx.x;
    int j = gid >> cshift;
    if (j >= k) return;
    int f = gid & ((1 << cshift) - 1);
    out[((size_t)perm[j] << cshift) + f] += xsmall[gid];
}

__global__ void concat_kernel(const float* __restrict__ a, const float* __restrict__ b,
                              float* cc, int n) {
    int gid = blockIdx.x * blockDim.x + threadIdx.x;
    if (gid >= n * CCAT) return;
    int i = gid >> 8, f = gid & 255;
    cc[gid] = (f < CH) ? a[(i << 7) + f] : b[(i << 7) + (f - CH)];
}

// ------------------------------ host driver --------------------------------

extern "C" void kernel_launch(void* const* d_in, const int* in_sizes, int n_in,
                              void* d_out, int out_size, void* d_ws, size_t ws_size,
                              hipStream_t stream) {
    (void)in_sizes; (void)n_in; (void)out_size; (void)ws_size;

    const float* xin = (const float*)d_in[0];
    const int*   ei  = (const int*)d_in[1];
    const float* Wd0 = (const float*)d_in[2];
    const float* Wd1 = (const float*)d_in[3];
    const float* Wd2 = (const float*)d_in[4];
    const float* pw0 = (const float*)d_in[5];
    const float* pw1 = (const float*)d_in[6];
    const float* Wu0 = (const float*)d_in[7];
    const float* Wu1 = (const float*)d_in[8];
    const float* Wu2 = (const float*)d_in[9];
    float* out = (float*)d_out;

    const int* er0 = ei;
    const int* ec0 = ei + NE;

    // ---- carve workspace ----
    char* base = (char*)d_ws;
    size_t cur = 0;
    auto carve = [&](size_t bytes) -> char* {
        char* p = base + cur;
        cur = (cur + bytes + 255) & ~(size_t)255;
        return p;
    };
    float* org  = (float*)carve((size_t)N0 * CH * 4);
    float* xs1  = (float*)carve((size_t)N1 * CH * 4);
    float* t0   = (float*)carve((size_t)N0 * CH * 4);
    float* t1   = (float*)carve((size_t)N0 * CH * 4);
    float* s0   = (float*)carve((size_t)N0 * CH * 4);   // hop scratch
    float* cc   = (float*)carve((size_t)N0 * CCAT * 4);
    float* ew0  = (float*)carve((size_t)NE * 4);
    int*   e1r  = (int*)  carve((size_t)NE * 4);
    int*   e1c  = (int*)  carve((size_t)NE * 4);
    float* ew1  = (float*)carve((size_t)NE * 4);
    int*   e2r  = (int*)  carve((size_t)NE * 4);
    int*   e2c  = (int*)  carve((size_t)NE * 4);
    float* ew2  = (float*)carve((size_t)NE * 4);
    // per-level CSR
    int*   rp0   = (int*)  carve((size_t)(N0 + 1) * 4);
    int*   ccol0 = (int*)  carve((size_t)NE * 4);
    float* cw0   = (float*)carve((size_t)NE * 4);
    float* dv0   = (float*)carve((size_t)N0 * 4);
    int*   rp1   = (int*)  carve((size_t)(N1 + 1) * 4);
    int*   ccol1 = (int*)  carve((size_t)NE * 4);
    float* cw1   = (float*)carve((size_t)NE * 4);
    float* dv1   = (float*)carve((size_t)N1 * 4);
    int*   rp2   = (int*)  carve((size_t)(N2 + 1) * 4);
    int*   ccol2 = (int*)  carve((size_t)NE * 4);
    float* cw2   = (float*)carve((size_t)NE * 4);
    float* dv2   = (float*)carve((size_t)N2 * 4);
    // build-time temporaries
    float* nrmT  = (float*)carve((size_t)NE * 4);
    int*   eidT  = (int*)  carve((size_t)NE * 4);
    int*   cntT  = (int*)  carve((size_t)N0 * 4);
    int*   posT  = (int*)  carve((size_t)N0 * 4);
    // pooling temporaries
    float* scr   = (float*)carve((size_t)N0 * 4);
    unsigned* key = (unsigned*)carve((size_t)N0 * 4);
    int*   nm    = (int*)  carve((size_t)N0 * 4);
    int*   perm0 = (int*)  carve((size_t)N1 * 4);
    int*   perm1 = (int*)  carve((size_t)N2 * 4);
    float* invn  = (float*)carve(64);
    unsigned* thr = (unsigned*)carve(64);
    int*   cnt   = (int*)carve(64);

    auto gemm = [&](const float* A, const float* W, float* O, int M, int K, int N) {
        dim3 g((unsigned)divup(M, 128), (unsigned)(N / 64));
        if (K <= 128)
            gemm_wmma_f32<128><<<g, 256, 0, stream>>>(A, W, O, M, K, N);
        else
            gemm_wmma_f32<256><<<g, 256, 0, stream>>>(A, W, O, M, K, N);
    };

    // build per-level normalization + deterministic CSR
    auto build_level = [&](const int* er, const int* ec, const float* ew, int n,
                           int* rp, int* ccol, float* cw, float* dv) {
        fill_f32<<<divup(n, 256), 256, 0, stream>>>(dv, 0.0f, n);
        deg_kernel<<<divup(NE, 256), 256, 0, stream>>>(er, ew, dv, NE);
        dinv_kernel<<<divup(n, 256), 256, 0, stream>>>(dv, n);
        norm_kernel<<<divup(NE, 256), 256, 0, stream>>>(er, ec, ew, dv, nrmT, NE);
        fill_i32<<<divup(n, 256), 256, 0, stream>>>(cntT, 0, n);
        rowcnt_kernel<<<divup(NE, 256), 256, 0, stream>>>(er, ew, cntT, NE);
        scan_kernel<<<1, 1024, 0, stream>>>(cntT, rp, n);
        fill_i32<<<divup(n, 256), 256, 0, stream>>>(posT, 0, n);
        csr_fill_kernel<<<divup(NE, 256), 256, 0, stream>>>(er, ew, rp, posT, eidT, NE);
        sort_rows_kernel<<<divup(n, 256), 256, 0, stream>>>(rp, eidT, n);
        csr_mat_kernel<<<divup(NE, 256), 256, 0, stream>>>(eidT, ec, nrmT, ccol, cw, rp, n, NE);
    };

    // outb = xa + Ah*xa + Ah^2*xa  (GCN 2-hop, deterministic gather)
    auto prop2 = [&](const float* xa, float* outb, float* h1buf, float* h2buf,
                     const int* rp, const int* ccol, const float* cw, const float* dv,
                     int n, int cshift) {
        int nc = n << cshift;
        int gthreads = n << (cshift - 2);
        copy_f32<<<divup(nc, 256), 256, 0, stream>>>(outb, xa, nc);
        hop_kernel<<<divup(gthreads, 256), 256, 0, stream>>>(rp, ccol, cw, dv, xa, h1buf, outb, n, cshift);
        hop_kernel<<<divup(gthreads, 256), 256, 0, stream>>>(rp, ccol, cw, dv, h1buf, h2buf, outb, n, cshift);
    };

    auto pool = [&](const float* x, const float* pw, int n, int k,
                    const int* er_in, const int* ec_in, const float* ew_in,
                    float* x2, int* perm, int* er_o, int* ec_o, float* ew_o) {
        pwnorm_kernel<<<1, 128, 0, stream>>>(pw, invn, CH);
        score_kernel<<<divup(n * 32, 256), 256, 0, stream>>>(x, pw, invn, scr, key, n, CH);
        topk_thresh_kernel<<<1, 1024, 0, stream>>>(key, n, k, thr);
        fill_i32<<<1, 32, 0, stream>>>(cnt, 0, 1);
        select_topk_kernel<<<divup(n, 256), 256, 0, stream>>>(key, thr, cnt, perm, n, k);
        fill_i32<<<divup(n, 256), 256, 0, stream>>>(nm, -1, n);
        build_nm_kernel<<<divup(k, 256), 256, 0, stream>>>(perm, nm, k);
        gather_gate_kernel<<<divup(k * CH, 256), 256, 0, stream>>>(x, perm, scr, x2, k, 7);
        relabel_kernel<<<divup(NE, 256), 256, 0, stream>>>(er_in, ec_in, ew_in, nm,
                                                          er_o, ec_o, ew_o, NE);
    };

    // ---- pipeline ----
    fill_f32<<<divup(NE, 256), 256, 0, stream>>>(ew0, 1.0f, NE);
    build_level(er0, ec0, ew0, N0, rp0, ccol0, cw0, dv0);

    // down conv 0 (full graph) -> org
    gemm(xin, Wd0, t1, N0, CH, CH);
    prop2(t1, org, s0, t0, rp0, ccol0, cw0, dv0, N0, 7);
    relu_f32<<<divup(N0 * CH, 256), 256, 0, stream>>>(org, N0 * CH);

    // pool 0 -> level-1 graph
    pool(org, pw0, N0, N1, er0, ec0, ew0, t0, perm0, e1r, e1c, ew1);
    build_level(e1r, e1c, ew1, N1, rp1, ccol1, cw1, dv1);

    // down conv 1 -> xs1
    gemm(t0, Wd1, t1, N1, CH, CH);
    prop2(t1, xs1, s0, t0, rp1, ccol1, cw1, dv1, N1, 7);
    relu_f32<<<divup(N1 * CH, 256), 256, 0, stream>>>(xs1, N1 * CH);

    // pool 1 -> level-2 graph
    pool(xs1, pw1, N1, N2, e1r, e1c, ew1, t0, perm1, e2r, e2c, ew2);
    build_level(e2r, e2c, ew2, N2, rp2, ccol2, cw2, dv2);

    // down conv 2 -> t0 (bottom features)
    gemm(t0, Wd2, t1, N2, CH, CH);
    prop2(t1, t0, s0, t1, rp2, ccol2, cw2, dv2, N2, 7);
    relu_f32<<<divup(N2 * CH, 256), 256, 0, stream>>>(t0, N2 * CH);

    // up 0: res = xs1, perm1
    copy_f32<<<divup(N1 * CH, 256), 256, 0, stream>>>(t1, xs1, N1 * CH);
    scatter_add_kernel<<<divup(N2 * CH, 256), 256, 0, stream>>>(t1, t0, perm1, N2, 7);
    gemm(t1, Wu0, t0, N1, CH, CH);
    prop2(t0, t1, s0, t0, rp1, ccol1, cw1, dv1, N1, 7);
    relu_f32<<<divup(N1 * CH, 256), 256, 0, stream>>>(t1, N1 * CH);

    // up 1: res = org, perm0 (no relu)
    copy_f32<<<divup(N0 * CH, 256), 256, 0, stream>>>(t0, org, N0 * CH);
    scatter_add_kernel<<<divup(N1 * CH, 256), 256, 0, stream>>>(t0, t1, perm0, N1, 7);
    gemm(t0, Wu1, t1, N0, CH, CH);
    prop2(t1, t0, s0, t1, rp0, ccol0, cw0, dv0, N0, 7);

    // final: concat [x, org] -> 256ch, conv with Wu2 -> 64ch into d_out
    concat_kernel<<<divup(N0 * CCAT, 256), 256, 0, stream>>>(t0, org, cc, N0);
    gemm(cc, Wu2, t1, N0, CCAT, COUT);
    prop2(t1, out, s0, cc, rp0, ccol0, cw0, dv0, N0, 6);
}